// RGAT_77953656422694
// MI455X (gfx1250) — compile-verified
//
#include <hip/hip_runtime.h>
#include <math.h>

// ---------------- problem constants (from reference) ----------------
#define NN    20000      // nodes
#define NE    640000     // edges
#define RR    90         // relations
#define INF_  128        // input feature dim
#define HID   64         // hidden
#define C1O   40         // conv1 per-relation out (H*O = 2*20)
#define C2O   64         // conv2 per-relation out (H*O = 2*32)
#define EPS   1e-16f

// ---------------- workspace layout (floats) ----------------
// XW region is reused: xw1 [90,20000,40] then xw2 [90,20000,64]
#define OFF_XW     ((size_t)0)            // 115,200,000 floats (max of both)
#define OFF_H      ((size_t)115200000)    // h  [N,64]        1,280,000
#define OFF_W1     ((size_t)116480000)    // w1 [90,64,40]      230,400
#define OFF_WE     ((size_t)116710400)    // we [16,2]               32
#define OFF_A1     ((size_t)116710432)    // alpha1 [E,4]     2,560,000
#define OFF_SMAX1  ((size_t)119270432)    // segmax1 [N*R*4]  7,200,000 (u32)
#define OFF_SSUM1  ((size_t)126470432)    // segsum1 [N*R*4]  7,200,000
#define OFF_H1     ((size_t)133670432)    // agg/h1 [N,80]    1,600,000
#define OFF_A2     ((size_t)135270432)    // alpha2 [E,2]     1,280,000
#define OFF_SMAX2  ((size_t)136550432)    // segmax2 [N*2]       40,000 (u32)
#define OFF_SSUM2  ((size_t)136590432)    // segsum2 [N*2]       40,000
#define OFF_H2     ((size_t)136630432)    // h2 [N,64]        1,280,000
// total: 137,910,432 floats = ~551.7 MB

typedef __attribute__((ext_vector_type(2))) float v2f;
typedef __attribute__((ext_vector_type(8))) float v8f;

__device__ __forceinline__ v8f wmma_f32(v2f a, v2f b, v8f c) {
  // V_WMMA_F32_16X16X4_F32 : D(16x16,f32) = A(16x4,f32) * B(4x16,f32) + C
  return __builtin_amdgcn_wmma_f32_16x16x4_f32(false, a, false, b, (short)0, c,
                                               false, false);
}

// monotonic float <-> uint mapping for atomicMax on floats
__device__ __forceinline__ unsigned fmap(float x) {
  unsigned u = __float_as_uint(x);
  return (u & 0x80000000u) ? ~u : (u ^ 0x80000000u);
}
__device__ __forceinline__ float funmap(unsigned u) {
  unsigned b = (u & 0x80000000u) ? (u ^ 0x80000000u) : ~u;
  return __uint_as_float(b);
}

// ---------------- utility: zero fill ----------------
__global__ void rgat_fill0(unsigned* p, long n) {
  long i = (long)blockIdx.x * blockDim.x + threadIdx.x;
  long st = (long)gridDim.x * blockDim.x;
  for (; i < n; i += st) p[i] = 0u;
}

// ---------------- small precompute: w1 = att1 @ basis1 ; we = lin_edge_w @ e2
__global__ void rgat_prep(const float* __restrict__ att1,
                          const float* __restrict__ basis1,
                          const float* __restrict__ lew,
                          const float* __restrict__ e2,
                          float* __restrict__ W1, float* __restrict__ WE) {
  int idx = blockIdx.x * blockDim.x + threadIdx.x;
  if (idx < RR * HID * C1O) {
    int r = idx / (HID * C1O);
    int io = idx % (HID * C1O);
    float acc = 0.f;
    for (int b = 0; b < 35; ++b)
      acc += att1[r * 35 + b] * basis1[(size_t)b * (HID * C1O) + io];
    W1[idx] = acc;
  }
  if (idx < 32) {
    int t = idx >> 1, c = idx & 1;
    float acc = 0.f;
    for (int f = 0; f < 64; ++f) acc += lew[t * 64 + f] * e2[f * 2 + c];
    WE[idx] = acc;
  }
}

// ---------------- h = x @ lin1_w   [20000,128]x[128,64] ----------------
__global__ void __launch_bounds__(128)
rgat_lin1(const float* __restrict__ x, const float* __restrict__ w,
          float* __restrict__ h) {
  __shared__ float As[16][INF_ + 1];
  __shared__ float Bs[INF_][HID];
  const int tid = threadIdx.x;
  const int m0 = blockIdx.x * 16;
  for (int i = tid; i < 16 * INF_; i += 128)
    As[i / INF_][i % INF_] = x[(size_t)(m0 + i / INF_) * INF_ + i % INF_];
  for (int i = tid; i < INF_ * HID; i += 128) Bs[i / HID][i % HID] = w[i];
  __syncthreads();
  const int lane = tid & 31, wv = tid >> 5;
  const int hl = lane >> 4, ll = lane & 15;
  const int nb = wv * 16;
  v8f c = {0.f, 0.f, 0.f, 0.f, 0.f, 0.f, 0.f, 0.f};
  for (int k = 0; k < INF_; k += 4) {
    int kb = k + hl * 2;
    v2f a = {As[ll][kb], As[ll][kb + 1]};
    v2f b = {Bs[kb][nb + ll], Bs[kb + 1][nb + ll]};
    c = wmma_f32(a, b, c);
  }
  for (int v = 0; v < 8; ++v) {
    int m = v + hl * 8;
    h[(size_t)(m0 + m) * HID + nb + ll] = c[v];
  }
}

// ---------------- xw1[r] = h @ w1[r]   90 x [20000,64]x[64,40] ----------------
__global__ void __launch_bounds__(96)
rgat_xw1(const float* __restrict__ h, const float* __restrict__ W1,
         float* __restrict__ xw) {
  __shared__ float As[16][HID + 1];
  __shared__ float Ws[HID][48];  // 40 cols + zero pad to 48
  const int tid = threadIdx.x;
  const int m0 = blockIdx.x * 16;
  const int r = blockIdx.y;
  for (int i = tid; i < 16 * HID; i += 96)
    As[i / HID][i % HID] = h[(size_t)(m0 + i / HID) * HID + i % HID];
  for (int i = tid; i < HID * 48; i += 96) {
    int row = i / 48, col = i % 48;
    Ws[row][col] = (col < C1O) ? W1[(size_t)r * HID * C1O + row * C1O + col] : 0.f;
  }
  __syncthreads();
  const int lane = tid & 31, wv = tid >> 5;
  const int hl = lane >> 4, ll = lane & 15;
  const int nb = wv * 16;
  v8f c = {0.f, 0.f, 0.f, 0.f, 0.f, 0.f, 0.f, 0.f};
  for (int k = 0; k < HID; k += 4) {
    int kb = k + hl * 2;
    v2f a = {As[ll][kb], As[ll][kb + 1]};
    v2f b = {Ws[kb][nb + ll], Ws[kb + 1][nb + ll]};
    c = wmma_f32(a, b, c);
  }
  if (nb + ll < C1O) {
    for (int v = 0; v < 8; ++v) {
      int m = v + hl * 8;
      xw[((size_t)r * NN + m0 + m) * C1O + nb + ll] = c[v];
    }
  }
}

// ---------------- xw2[r] = blockdiag(h1) @ w2[r] -> [90,20000,64] ----------------
__global__ void __launch_bounds__(128)
rgat_xw2(const float* __restrict__ h1, const float* __restrict__ w2,
         float* __restrict__ xw) {
  __shared__ float As[16][80 + 1];
  __shared__ float Ws[2][40][32];
  const int tid = threadIdx.x;
  const int m0 = blockIdx.x * 16;
  const int r = blockIdx.y;
  for (int i = tid; i < 16 * 80; i += 128)
    As[i / 80][i % 80] = h1[(size_t)(m0 + i / 80) * 80 + i % 80];
  for (int i = tid; i < 2 * 40 * 32; i += 128)
    ((float*)Ws)[i] = w2[(size_t)r * 2560 + i];
  __syncthreads();
  const int lane = tid & 31, wv = tid >> 5;
  const int hl = lane >> 4, ll = lane & 15;
  const int hh = wv >> 1;          // head (block of block-diagonal)
  const int nb = (wv & 1) * 16;    // n-tile within 32
  v8f c = {0.f, 0.f, 0.f, 0.f, 0.f, 0.f, 0.f, 0.f};
  for (int k = 0; k < 40; k += 4) {
    int kb = k + hl * 2;
    v2f a = {As[ll][hh * 40 + kb], As[ll][hh * 40 + kb + 1]};
    v2f b = {Ws[hh][kb][nb + ll], Ws[hh][kb + 1][nb + ll]};
    c = wmma_f32(a, b, c);
  }
  for (int v = 0; v < 8; ++v) {
    int m = v + hl * 8;
    xw[((size_t)r * NN + m0 + m) * C2O + hh * 32 + nb + ll] = c[v];
  }
}

// ---------------- conv1 edge kernels ----------------
__global__ void rgat_att1(const int* __restrict__ ei, const int* __restrict__ et,
                          const float* __restrict__ xw, const float* __restrict__ q1,
                          const float* __restrict__ k1, float* __restrict__ alpha,
                          unsigned* __restrict__ smax) {
  int e = blockIdx.x * blockDim.x + threadIdx.x;
  if (e >= NE) return;
  int s = ei[e], d = ei[NE + e], t = et[e];
  const float* pj = xw + ((size_t)t * NN + s) * C1O;
  const float* pi = xw + ((size_t)t * NN + d) * C1O;
  float qi[4] = {0, 0, 0, 0}, kj[4] = {0, 0, 0, 0};
  for (int o = 0; o < C1O; ++o) {
    float vi = pi[o], vj = pj[o];
    for (int c = 0; c < 4; ++c) {
      qi[c] += vi * q1[o * 4 + c];
      kj[c] += vj * k1[o * 4 + c];
    }
  }
  size_t seg = ((size_t)d * RR + t) * 4;
  for (int c = 0; c < 4; ++c) {
    float a = qi[c] * kj[c];
    alpha[(size_t)e * 4 + c] = a;
    atomicMax(&smax[seg + c], fmap(a));
  }
}

__global__ void rgat_exp1(const int* __restrict__ ei, const int* __restrict__ et,
                          float* __restrict__ alpha, const unsigned* __restrict__ smax,
                          float* __restrict__ ssum) {
  int e = blockIdx.x * blockDim.x + threadIdx.x;
  if (e >= NE) return;
  int d = ei[NE + e], t = et[e];
  size_t seg = ((size_t)d * RR + t) * 4;
  for (int c = 0; c < 4; ++c) {
    float m = funmap(smax[seg + c]);
    float ev = expf(alpha[(size_t)e * 4 + c] - m);
    alpha[(size_t)e * 4 + c] = ev;
    atomicAdd(&ssum[seg + c], ev);
  }
}

__global__ void rgat_msg1(const int* __restrict__ ei, const int* __restrict__ et,
                          const float* __restrict__ xw, const float* __restrict__ alpha,
                          const float* __restrict__ ssum, const float* __restrict__ wmod,
                          float* __restrict__ agg) {
  int e = blockIdx.x * blockDim.x + threadIdx.x;
  if (e >= NE) return;
  int s = ei[e], d = ei[NE + e], t = et[e];
  const float* pj = xw + ((size_t)t * NN + s) * C1O;
  size_t seg = ((size_t)d * RR + t) * 4;
  float a[4];
  for (int c = 0; c < 4; ++c)
    a[c] = alpha[(size_t)e * 4 + c] / (ssum[seg + c] + EPS);
  float* dst = agg + (size_t)d * 80;
  for (int hh = 0; hh < 2; ++hh)
    for (int o = 0; o < 20; ++o) {
      float oj = pj[hh * 20 + o];
      float wm = wmod[o];
      atomicAdd(&dst[(hh * 2 + 0) * 20 + o], oj * (a[hh * 2 + 0] + wm));
      atomicAdd(&dst[(hh * 2 + 1) * 20 + o], oj * (a[hh * 2 + 1] + wm));
    }
}

__global__ void rgat_h1relu(float* __restrict__ h1, const float* __restrict__ bias) {
  int i = blockIdx.x * blockDim.x + threadIdx.x;
  if (i >= NN * 80) return;
  float v = h1[i] + bias[i % 80];
  h1[i] = v > 0.f ? v : 0.f;
}

// ---------------- conv2 edge kernels ----------------
__global__ void rgat_att2(const int* __restrict__ ei, const int* __restrict__ et,
                          const float* __restrict__ xw, const float* __restrict__ q2,
                          const float* __restrict__ k2, const float* __restrict__ ea,
                          const float* __restrict__ we, float* __restrict__ alpha,
                          unsigned* __restrict__ smax) {
  int e = blockIdx.x * blockDim.x + threadIdx.x;
  if (e >= NE) return;
  int s = ei[e], d = ei[NE + e], t = et[e];
  const float* pj = xw + ((size_t)t * NN + s) * C2O;
  const float* pi = xw + ((size_t)t * NN + d) * C2O;
  float qi0 = 0, qi1 = 0, kj0 = 0, kj1 = 0;
  for (int f = 0; f < C2O; ++f) {
    float vi = pi[f], vj = pj[f];
    qi0 += vi * q2[f * 2];
    qi1 += vi * q2[f * 2 + 1];
    kj0 += vj * k2[f * 2];
    kj1 += vj * k2[f * 2 + 1];
  }
  float ae0 = 0, ae1 = 0;
  for (int tt = 0; tt < 16; ++tt) {
    float v = ea[(size_t)e * 16 + tt];
    ae0 += v * we[tt * 2];
    ae1 += v * we[tt * 2 + 1];
  }
  float a0 = qi0 + kj0 + ae0, a1 = qi1 + kj1 + ae1;
  a0 = a0 > 0.f ? a0 : 0.2f * a0;
  a1 = a1 > 0.f ? a1 : 0.2f * a1;
  alpha[(size_t)e * 2 + 0] = a0;
  alpha[(size_t)e * 2 + 1] = a1;
  atomicMax(&smax[d * 2 + 0], fmap(a0));
  atomicMax(&smax[d * 2 + 1], fmap(a1));
}

__global__ void rgat_exp2(const int* __restrict__ ei, float* __restrict__ alpha,
                          const unsigned* __restrict__ smax, float* __restrict__ ssum) {
  int e = blockIdx.x * blockDim.x + threadIdx.x;
  if (e >= NE) return;
  int d = ei[NE + e];
  for (int c = 0; c < 2; ++c) {
    float m = funmap(smax[d * 2 + c]);
    float ev = expf(alpha[(size_t)e * 2 + c] - m);
    alpha[(size_t)e * 2 + c] = ev;
    atomicAdd(&ssum[d * 2 + c], ev);
  }
}

__global__ void rgat_msg2(const int* __restrict__ ei, const int* __restrict__ et,
                          const float* __restrict__ xw, const float* __restrict__ alpha,
                          const float* __restrict__ ssum, float* __restrict__ h2) {
  int e = blockIdx.x * blockDim.x + threadIdx.x;
  if (e >= NE) return;
  int s = ei[e], d = ei[NE + e], t = et[e];
  const float* pj = xw + ((size_t)t * NN + s) * C2O;
  float c0 = alpha[(size_t)e * 2 + 0] / (ssum[d * 2 + 0] + EPS);
  float c1 = alpha[(size_t)e * 2 + 1] / (ssum[d * 2 + 1] + EPS);
  float* dst = h2 + (size_t)d * C2O;
  for (int f = 0; f < 32; ++f) atomicAdd(&dst[f], c0 * pj[f]);
  for (int f = 0; f < 32; ++f) atomicAdd(&dst[32 + f], c1 * pj[32 + f]);
}

// ---------------- final classifier + log_softmax ----------------
__global__ void rgat_out(const float* __restrict__ h2, const float* __restrict__ w,
                         const float* __restrict__ b, float* __restrict__ out) {
  int n = blockIdx.x * blockDim.x + threadIdx.x;
  if (n >= NN) return;
  float o[4] = {b[0], b[1], b[2], b[3]};
  const float* p = h2 + (size_t)n * C2O;
  for (int f = 0; f < C2O; ++f) {
    float v = p[f];
    for (int c = 0; c < 4; ++c) o[c] += v * w[f * 4 + c];
  }
  float m = fmaxf(fmaxf(o[0], o[1]), fmaxf(o[2], o[3]));
  float s = 0.f;
  for (int c = 0; c < 4; ++c) s += expf(o[c] - m);
  float l = logf(s);
  for (int c = 0; c < 4; ++c) out[(size_t)n * 4 + c] = o[c] - m - l;
}

// ---------------- launch ----------------
extern "C" void kernel_launch(void* const* d_in, const int* in_sizes, int n_in,
                              void* d_out, int out_size, void* d_ws, size_t ws_size,
                              hipStream_t stream) {
  (void)in_sizes; (void)n_in; (void)out_size; (void)ws_size;
  const float* x      = (const float*)d_in[0];
  const int*   ei     = (const int*)d_in[1];
  const int*   et     = (const int*)d_in[2];
  const float* ea     = (const float*)d_in[3];
  const float* lin1_w = (const float*)d_in[4];
  const float* att1   = (const float*)d_in[5];
  const float* basis1 = (const float*)d_in[6];
  const float* q1     = (const float*)d_in[7];
  const float* k1     = (const float*)d_in[8];
  const float* wmod1  = (const float*)d_in[9];
  const float* bias1  = (const float*)d_in[10];
  const float* w2     = (const float*)d_in[11];
  const float* q2     = (const float*)d_in[12];
  const float* k2     = (const float*)d_in[13];
  const float* lew    = (const float*)d_in[14];
  const float* e2     = (const float*)d_in[15];
  const float* lin2_w = (const float*)d_in[16];
  const float* lin2_b = (const float*)d_in[17];
  float* out = (float*)d_out;

  float*    ws    = (float*)d_ws;
  float*    XW    = ws + OFF_XW;
  float*    H     = ws + OFF_H;
  float*    W1    = ws + OFF_W1;
  float*    WE    = ws + OFF_WE;
  float*    A1    = ws + OFF_A1;
  unsigned* SMAX1 = (unsigned*)(ws + OFF_SMAX1);
  float*    SSUM1 = ws + OFF_SSUM1;
  float*    H1    = ws + OFF_H1;
  float*    A2    = ws + OFF_A2;
  unsigned* SMAX2 = (unsigned*)(ws + OFF_SMAX2);
  float*    SSUM2 = ws + OFF_SSUM2;
  float*    H2    = ws + OFF_H2;

  // zero init: [SMAX1|SSUM1|H1] and [SMAX2|SSUM2|H2] (contiguous regions)
  rgat_fill0<<<4096, 256, 0, stream>>>(SMAX1, 16000000L);
  rgat_fill0<<<512, 256, 0, stream>>>(SMAX2, 1360000L);

  // small precompute
  rgat_prep<<<900, 256, 0, stream>>>(att1, basis1, lew, e2, W1, WE);

  // dense GEMMs (WMMA f32)
  rgat_lin1<<<1250, 128, 0, stream>>>(x, lin1_w, H);
  rgat_xw1<<<dim3(1250, RR), 96, 0, stream>>>(H, W1, XW);

  // conv1 edge phase
  rgat_att1<<<2500, 256, 0, stream>>>(ei, et, XW, q1, k1, A1, SMAX1);
  rgat_exp1<<<2500, 256, 0, stream>>>(ei, et, A1, SMAX1, SSUM1);
  rgat_msg1<<<2500, 256, 0, stream>>>(ei, et, XW, A1, SSUM1, wmod1, H1);
  rgat_h1relu<<<6250, 256, 0, stream>>>(H1, bias1);

  // conv2 transform (reuses XW region; stream-ordered after conv1 edge phase)
  rgat_xw2<<<dim3(1250, RR), 128, 0, stream>>>(H1, w2, XW);

  // conv2 edge phase
  rgat_att2<<<2500, 256, 0, stream>>>(ei, et, XW, q2, k2, ea, WE, A2, SMAX2);
  rgat_exp2<<<2500, 256, 0, stream>>>(ei, A2, SMAX2, SSUM2);
  rgat_msg2<<<2500, 256, 0, stream>>>(ei, et, XW, A2, SSUM2, H2);

  // classifier + log_softmax
  rgat_out<<<79, 256, 0, stream>>>(H2, lin2_w, lin2_b, out);
}